// ModulatedConv2D_72619307041395
// MI455X (gfx1250) — compile-verified
//
#include <hip/hip_runtime.h>
#include <math.h>

// ---------------------------------------------------------------------------
// ModulatedConv2D (StyleGAN2) for MI455X / gfx1250.
//   B=8, H=W=256, CIN=COUT=64, K=3, STYLE_IN=512, fp32.
// Strategy: memory-bound implicit-GEMM conv using V_WMMA_F32_16X16X4_F32
// (full fp32 precision on the matrix pipe; ~11.5us HBM floor dominates).
// ---------------------------------------------------------------------------

#define BATCH    8
#define HH       256
#define WW       256
#define CIN      64
#define COUT     64
#define STYLE_IN 512
#define NTAPS    9
#define NKG      144            // 9 taps * 16 cin-groups of 4
#define WS_PER_B (NKG * COUT * 4)   // floats per batch in weight workspace

#define LIN_SCALE  0.04419417382415922f   // 1/sqrt(512)
#define CONV_SCALE 0.041666666666666664f  // 1/sqrt(64*9)

typedef __attribute__((ext_vector_type(2))) float v2f;
typedef __attribute__((ext_vector_type(8))) float v8f;

// ---------------------------------------------------------------------------
// Kernel 1: style modulation + demodulation; emit weights in WMMA-B layout:
//   wsW[b][kg][cout][klocal],  kg = (ky*3+kx)*16 + cin/4,  klocal = cin&3
// so a B fragment for (kg, coutTile) is one b64 load per lane:
//   lane<16  -> { K0, K1 } = klocal {0,1} of cout = tile+lane
//   lane>=16 -> { K2, K3 } = klocal {2,3} of cout = tile+lane-16
// ---------------------------------------------------------------------------
__global__ __launch_bounds__(256)
void modconv_weights_kernel(const float* __restrict__ style,       // [8,512]
                            const float* __restrict__ mod_weight,  // [512,64]
                            const float* __restrict__ mod_bias,    // [64]
                            const float* __restrict__ weight,      // [64,64,3,3]
                            float* __restrict__ wsW)               // [8,144,64,4]
{
    const int b = blockIdx.x;
    const int t = threadIdx.x;

    __shared__ float s_mod[CIN];
    __shared__ float s_part[256];
    __shared__ float s_demod[COUT];

    // s = style @ (W * LIN_SCALE) + bias   (one thread per cin)
    if (t < CIN) {
        float acc = mod_bias[t];
        const float* st = style + b * STYLE_IN;
        for (int k = 0; k < STYLE_IN; ++k)
            acc += st[k] * (mod_weight[k * CIN + t] * LIN_SCALE);
        s_mod[t] = acc;
    }
    __syncthreads();

    // sum of squares of modulated weights per cout (4 threads per cout)
    {
        const int cout = t >> 2;
        const int part = t & 3;
        float ss = 0.f;
        const float* wco = weight + (size_t)cout * CIN * NTAPS;
        for (int cin = part * 16; cin < part * 16 + 16; ++cin) {
            const float sm = CONV_SCALE * s_mod[cin];
            const float* wp = wco + cin * NTAPS;
            #pragma unroll
            for (int tap = 0; tap < NTAPS; ++tap) {
                const float v = wp[tap] * sm;
                ss += v * v;
            }
        }
        s_part[t] = ss;
    }
    __syncthreads();
    if ((t & 3) == 0) {
        const float ss = s_part[t] + s_part[t + 1] + s_part[t + 2] + s_part[t + 3];
        s_demod[t >> 2] = rsqrtf(ss + 1e-8f);
    }
    __syncthreads();

    // final demodulated weights in B-operand order
    float* wsb = wsW + (size_t)b * WS_PER_B;
    for (int lin = t; lin < WS_PER_B; lin += 256) {
        const int klocal = lin & 3;
        const int cout   = (lin >> 2) & 63;
        const int kg     = lin >> 8;
        const int cin    = ((kg & 15) << 2) + klocal;
        const int tap    = kg >> 4;          // ky*3 + kx
        const float v = CONV_SCALE * weight[((size_t)cout * CIN + cin) * NTAPS + tap]
                        * s_mod[cin] * s_demod[cout];
        wsb[lin] = v;
    }
}

// ---------------------------------------------------------------------------
// Kernel 2: implicit-GEMM 3x3 SAME conv, one batch-row-pair per block.
//   grid = B * H/2 blocks, 256 threads (8 waves).
//   wave -> (row y, 64-pixel segment); 4 M-tiles x 4 N-tiles of 16x16 fp32.
//   K = 576 reduced in 144 steps of 4 via v_wmma_f32_16x16x4_f32.
// ---------------------------------------------------------------------------
__global__ __launch_bounds__(256)
void modconv_conv_kernel(const float* __restrict__ in,   // [8,256,256,64] NHWC
                         const float* __restrict__ wsW,  // [8,144,64,4]
                         float* __restrict__ out)        // [8,256,256,64] NHWC
{
    const int blk  = blockIdx.x;
    const int b    = blk >> 7;          // / (H/2)
    const int yp   = blk & 127;
    const int wave = threadIdx.x >> 5;
    const int lane = threadIdx.x & 31;

    const int y    = yp * 2 + (wave >> 2);
    const int x0   = (wave & 3) * 64;
    const int m    = lane & 15;         // N / M-row index within a 16-tile
    const int half = lane >> 4;         // K-half selector (0: K0/K1, 1: K2/K3)

    v8f acc[4][4];
    #pragma unroll
    for (int mt = 0; mt < 4; ++mt)
        #pragma unroll
        for (int nt = 0; nt < 4; ++nt)
            acc[mt][nt] = (v8f){0.f, 0.f, 0.f, 0.f, 0.f, 0.f, 0.f, 0.f};

    const float* in_b    = in  + (size_t)b * HH * WW * CIN;
    const float* wb_base = wsW + (size_t)b * WS_PER_B;

    for (int ky = 0; ky < 3; ++ky) {
        const int yy = y + ky - 1;
        if (yy < 0 || yy >= HH) continue;            // wave-uniform skip
        const float* in_row = in_b + (size_t)yy * WW * CIN;

        #pragma unroll
        for (int kx = 0; kx < 3; ++kx) {
            const int tap = ky * 3 + kx;
            const float* wb_tap = wb_base + (size_t)tap * 16 * COUT * 4;

            for (int g = 0; g < 16; ++g) {           // cin group of 4
                const int cin0 = g * 4;

                // A fragments: per-lane predicated NHWC channel-pair loads
                v2f afrag[4];
                #pragma unroll
                for (int mt = 0; mt < 4; ++mt) {
                    const int xx = x0 + mt * 16 + m + kx - 1;
                    v2f a = (v2f){0.f, 0.f};
                    if (xx >= 0 && xx < WW)
                        a = *(const v2f*)(in_row + (size_t)xx * CIN + cin0 + 2 * half);
                    afrag[mt] = a;
                }

                // B fragments: L2-hot weights, one b64 load per lane
                const float* wg = wb_tap + (size_t)g * COUT * 4 + 2 * half;
                v2f bfrag[4];
                #pragma unroll
                for (int nt = 0; nt < 4; ++nt)
                    bfrag[nt] = *(const v2f*)(wg + (nt * 16 + m) * 4);

                // 16 fp32 WMMAs: acc[mt][nt] += A[mt] x B[nt]
                #pragma unroll
                for (int mt = 0; mt < 4; ++mt)
                    #pragma unroll
                    for (int nt = 0; nt < 4; ++nt)
                        acc[mt][nt] = __builtin_amdgcn_wmma_f32_16x16x4_f32(
                            false, afrag[mt], false, bfrag[nt],
                            (short)0, acc[mt][nt], false, false);
            }
        }
    }

    // Epilogue: C/D layout -> VGPR r, lanes 0-15: M=r, N=lane;
    //                        lanes 16-31: M=r+8, N=lane-16
    float* out_row = out + (((size_t)b * HH + y) * WW) * COUT;
    #pragma unroll
    for (int mt = 0; mt < 4; ++mt) {
        #pragma unroll
        for (int r = 0; r < 8; ++r) {
            const int x = x0 + mt * 16 + r + 8 * half;
            float* op = out_row + (size_t)x * COUT;
            #pragma unroll
            for (int nt = 0; nt < 4; ++nt)
                op[nt * 16 + m] = acc[mt][nt][r];
        }
    }
}

// ---------------------------------------------------------------------------
extern "C" void kernel_launch(void* const* d_in, const int* in_sizes, int n_in,
                              void* d_out, int out_size, void* d_ws, size_t ws_size,
                              hipStream_t stream) {
    const float* inputs     = (const float*)d_in[0];  // [8,256,256,64]
    const float* style      = (const float*)d_in[1];  // [8,512]
    const float* mod_weight = (const float*)d_in[2];  // [512,64]
    const float* mod_bias   = (const float*)d_in[3];  // [64]
    const float* weight     = (const float*)d_in[4];  // [1,64,64,3,3]
    float* out = (float*)d_out;                       // [8,256,256,64]
    float* wsW = (float*)d_ws;                        // needs 8*144*64*4*4 = 1.18 MB

    modconv_weights_kernel<<<BATCH, 256, 0, stream>>>(style, mod_weight, mod_bias,
                                                      weight, wsW);
    modconv_conv_kernel<<<BATCH * (HH / 2), 256, 0, stream>>>(inputs, wsW, out);
}